// gcnLayer_28243704939341
// MI455X (gfx1250) — compile-verified
//
#include <hip/hip_runtime.h>

#define DIN  256
#define DOUT 128

typedef float v2f __attribute__((ext_vector_type(2)));
typedef float v8f __attribute__((ext_vector_type(8)));
typedef unsigned int u32x4 __attribute__((ext_vector_type(4)));
typedef int i32x4 __attribute__((ext_vector_type(4)));
typedef int i32x8 __attribute__((ext_vector_type(8)));

// ---------------------------------------------------------------------------
// 1. zero accumulator (N*DOUT) and degree (N)
// ---------------------------------------------------------------------------
__global__ void zero_kernel(float* __restrict__ acc, float* __restrict__ deg,
                            int n_acc, int n_deg) {
    int i = blockIdx.x * blockDim.x + threadIdx.x;
    if (i < n_acc) acc[i] = 0.0f;
    if (i < n_deg) deg[i] = 0.0f;
}

// ---------------------------------------------------------------------------
// 2. degree over destinations: deg[dst[e]] += ew[e]  (hw f32 atomic)
// ---------------------------------------------------------------------------
__global__ void degree_kernel(const int* __restrict__ dst,
                              const float* __restrict__ ew,
                              float* __restrict__ deg, int E) {
    int e = blockIdx.x * blockDim.x + threadIdx.x;
    if (e < E) unsafeAtomicAdd(&deg[dst[e]], ew[e]);
}

// ---------------------------------------------------------------------------
// 3. dinv = rsqrt(deg + 1)   (self-loop weight 1 folded in; deg>=1 so >0)
// ---------------------------------------------------------------------------
__global__ void dinv_kernel(const float* __restrict__ deg,
                            float* __restrict__ dinv, int N) {
    int i = blockIdx.x * blockDim.x + threadIdx.x;
    if (i < N) {
        float d = deg[i] + 1.0f;
        dinv[i] = (d > 0.0f) ? rsqrtf(d) : 0.0f;
    }
}

// ---------------------------------------------------------------------------
// 4. x = feat @ W via V_WMMA_F32_16X16X4_F32 (exact fp32)
//    block = 256 threads = 8 wave32; block computes rows [16*bid, 16*bid+16)
//    A-tile (16 rows x 256 cols, 16 KB) staged into LDS by the Tensor Data
//    Mover: wave 0 issues one 2D tensor_load_to_lds descriptor, waits
//    TENSORcnt==0, then the workgroup barrier releases the other 7 waves.
//    Wave w computes the 16x16 output tile at columns [16w, 16w+16).
//    A (16x4 f32): lane l(0..15): v0=A[l][k0], v1=A[l][k0+1], k0=(l>>4)*2
//    B (4x16 f32): lane l: v0=B[k0][l&15], v1=B[k0+1][l&15]
//    C/D (16x16 f32, 8 VGPR): vgpr r, lanes 0-15 -> M=r, lanes 16-31 -> M=r+8
// ---------------------------------------------------------------------------
__launch_bounds__(256, 2)
__global__ void gemm_wmma_kernel(const float* __restrict__ feat,
                                 const float* __restrict__ W,
                                 float* __restrict__ x, int N) {
    __shared__ float Atile[16 * DIN];                 // 16 KB of the 320 KB LDS

    const int row0 = blockIdx.x * 16;

    if (threadIdx.x < 32) {
        // --- Tensor DMA descriptor (D#), CDNA5 ISA sect 8 ---
        // generic pointer to LDS: low 32 bits are the LDS byte address
        unsigned lds_addr = (unsigned)(unsigned long long)&Atile[0];
        unsigned long long ga =
            (unsigned long long)(feat + (size_t)row0 * DIN);
        unsigned rows = (unsigned)(N - row0);         // clamps OOB rows to 0

        u32x4 g0;
        g0[0] = 1u;                                   // count=1, user mode
        g0[1] = lds_addr;                             // lds_addr[31:0]
        g0[2] = (unsigned)(ga & 0xFFFFFFFFull);       // global_addr[31:0]
        g0[3] = (unsigned)((ga >> 32) & 0x01FFFFFFull)// global_addr[56:32]
              | (2u << 30);                           // type=2 ("image")

        i32x8 g1;
        g1[0] = (int)(2u << 16);                      // data_size=2 -> 4 bytes
        g1[1] = (int)((DIN & 0xFFFF) << 16);          // tensor_dim0[15:0]
        g1[2] = (int)((DIN >> 16) |                   // tensor_dim0[31:16]
                      ((rows & 0xFFFFu) << 16));      // tensor_dim1[15:0]
        g1[3] = (int)(((rows >> 16) & 0xFFFFu) |      // tensor_dim1[31:16]
                      (DIN << 16));                   // tile_dim0 = 256
        g1[4] = 16;                                   // tile_dim1=16, tile_dim2=0
        g1[5] = DIN;                                  // tensor_dim0_stride lo32
        g1[6] = 0;                                    // stride0 hi | stride1 lo
        g1[7] = 0;                                    // stride1 hi

        i32x4 z = {0, 0, 0, 0};
#if __clang_major__ >= 23
        i32x8 z8 = {0, 0, 0, 0, 0, 0, 0, 0};
        __builtin_amdgcn_tensor_load_to_lds(g0, g1, z, z, z8, 0);
#else
        __builtin_amdgcn_tensor_load_to_lds(g0, g1, z, z, 0);
#endif
        __builtin_amdgcn_s_wait_tensorcnt(0);
    }
    __syncthreads();

    const int wave  = threadIdx.x >> 5;               // 0..7 -> column tile
    const int lane  = threadIdx.x & 31;
    const int half  = lane >> 4;                      // 0 or 1
    const int mrow  = lane & 15;                      // A row / D column index
    const int khalf = half << 1;                      // 0 or 2 within K=4 slab
    const int col   = (wave << 4) + mrow;             // global output column

    v8f c = {};
    #pragma unroll 4
    for (int k = 0; k < DIN; k += 4) {
        v2f a, b;
        const float2 av = *(const float2*)(&Atile[mrow * DIN + k + khalf]);
        a.x = av.x;
        a.y = av.y;
        b.x = W[(k + khalf) * DOUT + col];
        b.y = W[(k + khalf + 1) * DOUT + col];
        // D = A*B + C ; 8 args: (neg_a, A, neg_b, B, c_mod, C, reuse_a, reuse_b)
        c = __builtin_amdgcn_wmma_f32_16x16x4_f32(false, a, false, b,
                                                  (short)0, c, false, false);
    }

    const int mbase = half * 8;
    float* xp = x + (size_t)(row0 + mbase) * DOUT + col;
    if (row0 + 16 <= N) {                             // uniform: full tile
        #pragma unroll
        for (int r = 0; r < 8; ++r) xp[(size_t)r * DOUT] = c[r];
    } else {                                          // ragged tail (unused for N=50000)
        #pragma unroll
        for (int r = 0; r < 8; ++r)
            if (row0 + mbase + r < N) xp[(size_t)r * DOUT] = c[r];
    }
}

// ---------------------------------------------------------------------------
// 5. edge scatter: 32 lanes per edge, 4 features per lane (float4 gather)
//    acc[dst] += x[src] * (dinv[src]*ew*dinv[dst])
// ---------------------------------------------------------------------------
__global__ void scatter_kernel(const float* __restrict__ x,
                               const int* __restrict__ src,
                               const int* __restrict__ dst,
                               const float* __restrict__ ew,
                               const float* __restrict__ dinv,
                               float* __restrict__ acc, int E) {
    long long tid = (long long)blockIdx.x * blockDim.x + threadIdx.x;
    int e = (int)(tid >> 5);
    if (e >= E) return;
    int lane = (int)(tid & 31);

    int s = src[e];
    int d = dst[e];
    float norm = dinv[s] * ew[e] * dinv[d];

    const float4 v = *(const float4*)(x + (size_t)s * DOUT + lane * 4);
    float* p = acc + (size_t)d * DOUT + lane * 4;
    unsafeAtomicAdd(p + 0, v.x * norm);
    unsafeAtomicAdd(p + 1, v.y * norm);
    unsafeAtomicAdd(p + 2, v.z * norm);
    unsafeAtomicAdd(p + 3, v.w * norm);
}

// ---------------------------------------------------------------------------
// 6. finalize: out = tanh(acc + x*dinv^2 (self loop) + b)
// ---------------------------------------------------------------------------
__global__ void finalize_kernel(const float* __restrict__ acc,
                                const float* __restrict__ x,
                                const float* __restrict__ dinv,
                                const float* __restrict__ b,
                                float* __restrict__ out, int total) {
    int i = blockIdx.x * blockDim.x + threadIdx.x;
    if (i >= total) return;
    int row = i >> 7;            // /DOUT
    int colf = i & (DOUT - 1);
    float di = dinv[row];
    float v = acc[i] + x[i] * di * di + b[colf];
    out[i] = tanhf(v);
}

// ---------------------------------------------------------------------------
extern "C" void kernel_launch(void* const* d_in, const int* in_sizes, int n_in,
                              void* d_out, int out_size, void* d_ws, size_t ws_size,
                              hipStream_t stream) {
    const float* feat = (const float*)d_in[0];
    const int*   ei   = (const int*)d_in[1];
    const float* ew   = (const float*)d_in[2];
    const float* W    = (const float*)d_in[3];
    const float* bias = (const float*)d_in[4];
    float* out = (float*)d_out;

    const int N = in_sizes[0] / DIN;        // 50000
    const int E = in_sizes[1] / 2;          // 1600000
    const int* src = ei;
    const int* dst = ei + E;

    // workspace layout (floats): deg[N] | dinv[N] | x[N*DOUT] | acc[N*DOUT]
    float* deg  = (float*)d_ws;
    float* dinv = deg + N;
    float* x    = dinv + N;
    float* acc  = x + (size_t)N * DOUT;

    const int total = N * DOUT;

    zero_kernel<<<(total + 255) / 256, 256, 0, stream>>>(acc, deg, total, N);

    degree_kernel<<<(E + 255) / 256, 256, 0, stream>>>(dst, ew, deg, E);

    dinv_kernel<<<(N + 255) / 256, 256, 0, stream>>>(deg, dinv, N);

    gemm_wmma_kernel<<<(N + 15) / 16, 256, 0, stream>>>(feat, W, x, N);

    long long sthreads = (long long)E * 32;
    scatter_kernel<<<(int)((sthreads + 255) / 256), 256, 0, stream>>>(
        x, src, dst, ew, dinv, acc, E);

    finalize_kernel<<<(total + 255) / 256, 256, 0, stream>>>(acc, x, dinv, bias,
                                                             out, total);
}